// CRF_18786186952978
// MI455X (gfx1250) — compile-verified
//
#include <hip/hip_runtime.h>
#include <hip/hip_bf16.h>

// Problem constants (match reference)
#define BB   128
#define SS   512
#define TT   36
#define START_TAG 34
#define END_TAG   35

#define LOG2E_F 1.4426950408889634f
#define LN2_F   0.6931471805599453f

typedef __attribute__((ext_vector_type(2))) float v2f;
typedef __attribute__((ext_vector_type(8))) float v8f;

// Pure-VALU half-swap within wave32: lane i <-> lane i^16 (v_permlanex16_b32).
__device__ __forceinline__ float half_swap(float x) {
  int r = __builtin_amdgcn_permlanex16(__float_as_int(x), __float_as_int(x),
                                       0x76543210, 0xfedcba98, false, false);
  return __int_as_float(r);
}

// K-permutation (compile-time folded): low-half-resident tags feed low-half B
// slots, high-half tags feed high-half slots; tags 34/35 contribute exactly 0
// in f32 and are dropped; 2 zero pads fill the remaining high slots.
constexpr int LTAG[18] = {0,1,2,3,4,5,6,7,16,17,18,19,20,21,22,23,32,33};
constexpr int HTAG[16] = {8,9,10,11,12,13,14,15,24,25,26,27,28,29,30,31};
// valid row count per D-tile (tile 2 rows 4..7 are identically 0)
constexpr int NROWS[3] = {8, 8, 4};

// ---------------------------------------------------------------------------
// Forward (partition) kernel. grid = 8 blocks x 1 wave.
// Exp-domain scan with exact power-of-2 renormalization every 8 steps:
//   alpha[b,j] = V[b,j] * 2^C[b]
//   per step:  V' = (E^T @ V) * exp2(feat*log2e)   (27x V_WMMA_F32_16X16X4_F32)
// The exp2(feat) factor depends only on loads -> off the serial critical path.
// ---------------------------------------------------------------------------
__global__ void
__attribute__((amdgpu_flat_work_group_size(32, 32), amdgpu_waves_per_eu(1)))
crf_forward_kernel(
    const float* __restrict__ feats,   // (B,S,T)
    const float* __restrict__ trans,   // (T,T)
    const int*   __restrict__ mask,    // (B,S)
    float*       __restrict__ ws)      // ws[0] += forward_score partial
{
  const int lane = threadIdx.x & 31;
  const int half = lane >> 4;
  const int l16  = lane & 15;
  const int b    = blockIdx.x * 16 + l16;

  // ---- constant A-matrix: E^T[j, slot] = exp(trans[tag(slot)][j]) ----
  float Am[3][9][2];
  #pragma unroll
  for (int mt = 0; mt < 3; ++mt) {
    const int j = 16 * mt + l16;
    #pragma unroll
    for (int c = 0; c < 9; ++c) {
      #pragma unroll
      for (int rr = 0; rr < 2; ++rr) {
        const int idx  = 2 * c + rr;
        const int tagL = LTAG[idx];
        const int tagH = (idx < 16) ? HTAG[idx] : -1;
        float vL = 0.0f, vH = 0.0f;
        if (j < TT) {
          vL = __builtin_amdgcn_exp2f(LOG2E_F * trans[tagL * TT + j]);
          if (tagH >= 0)
            vH = __builtin_amdgcn_exp2f(LOG2E_F * trans[tagH * TT + j]);
        }
        Am[mt][c][rr] = half ? vH : vL;
      }
    }
  }

  // ---- init: q0 = log2(alpha0);  C = max(q0);  V = 2^(q0 - C) ----
  float V[3][8];
  float C;
  {
    float q[3][8];
    #pragma unroll
    for (int mt = 0; mt < 3; ++mt) {
      #pragma unroll
      for (int r = 0; r < NROWS[mt]; ++r) {
        const int j = 16 * mt + 8 * half + r;
        q[mt][r] = (j < TT)
                     ? LOG2E_F * (feats[(size_t)b * SS * TT + j] + trans[START_TAG * TT + j])
                     : -__builtin_inff();
      }
    }
    float mx = q[0][0];
    #pragma unroll
    for (int mt = 0; mt < 3; ++mt)
      #pragma unroll
      for (int r = 0; r < NROWS[mt]; ++r)
        mx = fmaxf(mx, q[mt][r]);
    mx = fmaxf(mx, half_swap(mx));
    C = mx;
    #pragma unroll
    for (int mt = 0; mt < 3; ++mt)
      #pragma unroll
      for (int r = 0; r < NROWS[mt]; ++r)
        V[mt][r] = __builtin_amdgcn_exp2f(q[mt][r] - mx);
  }

  // per-lane streaming pointers (step s = 1)
  const float* fptr = feats + ((size_t)b * SS + 1) * TT;
  const int*   mptr = mask + b * SS + 1;
  const float* pfend = feats + ((size_t)b * SS + (SS - 1)) * TT;
  const float* pfptr = fptr + 8 * TT + (half ? 32 : 0);

  // --- one scan step (exp domain) ---
  auto step_fn = [&]() {
    const int mk = *mptr;  mptr += 1;

    {
      const float* pp = pfptr < pfend ? pfptr : pfend;
      __builtin_prefetch(pp, 0, 1);
      pfptr += TT;
    }

    // feats -> F = 2^(feat*log2e) (independent of V: runs under the WMMAs)
    float F[3][8];
    {
      const float4* f4 = (const float4*)(fptr + 8 * half);
      float4 a0 = f4[0];
      float4 a1 = f4[1];
      const float4* g4 = (const float4*)(fptr + 16 + 8 * half);
      float4 b0 = g4[0];
      float4 b1 = g4[1];
      float4 c0 = make_float4(0.f, 0.f, 0.f, 0.f);
      if (half == 0) c0 = *(const float4*)(fptr + 32);
      float fj[3][8];
      fj[0][0]=a0.x; fj[0][1]=a0.y; fj[0][2]=a0.z; fj[0][3]=a0.w;
      fj[0][4]=a1.x; fj[0][5]=a1.y; fj[0][6]=a1.z; fj[0][7]=a1.w;
      fj[1][0]=b0.x; fj[1][1]=b0.y; fj[1][2]=b0.z; fj[1][3]=b0.w;
      fj[1][4]=b1.x; fj[1][5]=b1.y; fj[1][6]=b1.z; fj[1][7]=b1.w;
      fj[2][0]=c0.x; fj[2][1]=c0.y; fj[2][2]=c0.z; fj[2][3]=c0.w;
      #pragma unroll
      for (int mt = 0; mt < 3; ++mt)
        #pragma unroll
        for (int r = 0; r < NROWS[mt]; ++r)
          F[mt][r] = __builtin_amdgcn_exp2f(fj[mt][r] * LOG2E_F);
    }
    fptr += TT;

    // B-layout build from V: per-lane select only
    float Bm[9][2];
    #pragma unroll
    for (int c = 0; c < 9; ++c) {
      #pragma unroll
      for (int rr = 0; rr < 2; ++rr) {
        const int idx  = 2 * c + rr;
        const int tagL = LTAG[idx];
        const float vL = V[tagL >> 4][tagL & 15];
        const float vH = (idx < 16)
                           ? V[HTAG[idx] >> 4][(HTAG[idx] & 15) - 8]
                           : 0.0f;
        Bm[c][rr] = half ? vH : vL;
      }
    }

    // Y^T = E^T @ V : 3 M-tiles x 9 K-chunks
    v8f acc[3];
    #pragma unroll
    for (int mt = 0; mt < 3; ++mt)
      acc[mt] = (v8f){0.f, 0.f, 0.f, 0.f, 0.f, 0.f, 0.f, 0.f};
    #pragma unroll
    for (int c = 0; c < 9; ++c) {
      v2f bv;  bv.x = Bm[c][0];  bv.y = Bm[c][1];
      #pragma unroll
      for (int mt = 0; mt < 3; ++mt) {
        v2f av;  av.x = Am[mt][c][0];  av.y = Am[mt][c][1];
        acc[mt] = __builtin_amdgcn_wmma_f32_16x16x4_f32(
            false, av, false, bv, (short)0, acc[mt], false, false);
      }
    }

    // V' = Y * F, masked per batch row
    #pragma unroll
    for (int mt = 0; mt < 3; ++mt) {
      #pragma unroll
      for (int r = 0; r < NROWS[mt]; ++r) {
        const float nv = acc[mt][r] * F[mt][r];
        V[mt][r] = mk ? nv : V[mt][r];
      }
    }
  };

  // --- exact power-of-2 renormalization ---
  auto renorm_fn = [&]() {
    float mx = V[0][0];
    #pragma unroll
    for (int mt = 0; mt < 3; ++mt)
      #pragma unroll
      for (int r = 0; r < NROWS[mt]; ++r)
        mx = fmaxf(mx, V[mt][r]);
    mx = fmaxf(mx, half_swap(mx));
    const int ebits = (__float_as_int(mx) >> 23) & 0xff;
    const float scale = __int_as_float((254 - ebits) << 23);   // 2^(127-e)
    C += (float)(ebits - 127);
    #pragma unroll
    for (int mt = 0; mt < 3; ++mt)
      #pragma unroll
      for (int r = 0; r < NROWS[mt]; ++r)
        V[mt][r] *= scale;
  };

  // 511 steps total: 63 blocks of 8 + tail of 7, renorm between blocks
  for (int blk = 0; blk < 63; ++blk) {
    if (blk) renorm_fn();
    #pragma unroll
    for (int t = 0; t < 8; ++t) step_fn();
  }
  renorm_fn();
  #pragma unroll
  for (int t = 0; t < 7; ++t) step_fn();

  // ---- final: forward_b = ln2 * (C + log2(sum_i V_i * exp(trans[i][END])))
  {
    float Bm[9][2];
    #pragma unroll
    for (int c = 0; c < 9; ++c) {
      #pragma unroll
      for (int rr = 0; rr < 2; ++rr) {
        const int idx  = 2 * c + rr;
        const int tagL = LTAG[idx];
        const float vL = V[tagL >> 4][tagL & 15];
        const float vH = (idx < 16)
                           ? V[HTAG[idx] >> 4][(HTAG[idx] & 15) - 8]
                           : 0.0f;
        Bm[c][rr] = half ? vH : vL;
      }
    }

    // only tile mt=2 needed (END=35 -> reg 3, half-0 lanes)
    v8f acc2 = (v8f){0.f, 0.f, 0.f, 0.f, 0.f, 0.f, 0.f, 0.f};
    #pragma unroll
    for (int c = 0; c < 9; ++c) {
      v2f av;  av.x = Am[2][c][0];  av.y = Am[2][c][1];
      v2f bv;  bv.x = Bm[c][0];     bv.y = Bm[c][1];
      acc2 = __builtin_amdgcn_wmma_f32_16x16x4_f32(
          false, av, false, bv, (short)0, acc2, false, false);
    }

    const float fe = LN2_F * (C + __builtin_amdgcn_logf(acc2[3]));
    float contrib = (half == 0) ? fe : 0.0f;
    #pragma unroll
    for (int off = 1; off < 32; off <<= 1)
      contrib += __shfl_xor(contrib, off, 32);
    if (lane == 0) atomicAdd(&ws[0], contrib);
  }
}

// ---------------------------------------------------------------------------
// Gold-path score: one block per batch element.
// ---------------------------------------------------------------------------
__global__ __launch_bounds__(256) void crf_gold_kernel(
    const float* __restrict__ feats, const float* __restrict__ trans,
    const int* __restrict__ mask, const int* __restrict__ tags,
    float* __restrict__ ws)            // ws[1] += gold_score partial
{
  const int b   = blockIdx.x;
  const int tid = threadIdx.x;

  float acc = 0.0f;
  int   len = 0;
  for (int s = tid; s < SS; s += 256) {
    const int mk   = mask[b * SS + s];
    const int cur  = tags[b * SS + s];
    const int prev = (s == 0) ? START_TAG : tags[b * SS + s - 1];
    const float v  = feats[((size_t)b * SS + s) * TT + cur] + trans[prev * TT + cur];
    acc += mk ? v : 0.0f;
    len += mk;
  }

  __shared__ float sf[256];
  __shared__ int   si[256];
  sf[tid] = acc;
  si[tid] = len;
  __syncthreads();
  #pragma unroll
  for (int off = 128; off > 0; off >>= 1) {
    if (tid < off) { sf[tid] += sf[tid + off]; si[tid] += si[tid + off]; }
    __syncthreads();
  }
  if (tid == 0) {
    const int L      = si[0];
    const int end_id = tags[b * SS + (L - 1)];
    const float g    = sf[0] + trans[end_id * TT + END_TAG];
    atomicAdd(&ws[1], g);
  }
}

__global__ void crf_finalize_kernel(const float* __restrict__ ws,
                                    float* __restrict__ out)
{
  out[0] = (ws[0] - ws[1]) * (1.0f / (float)BB);
}

extern "C" void kernel_launch(void* const* d_in, const int* in_sizes, int n_in,
                              void* d_out, int out_size, void* d_ws, size_t ws_size,
                              hipStream_t stream) {
  const float* feats = (const float*)d_in[0];   // (B,S,T) f32
  const float* trans = (const float*)d_in[1];   // (T,T)   f32
  const int*   mask  = (const int*)d_in[2];     // (B,S)   i32
  const int*   tags  = (const int*)d_in[3];     // (B,S)   i32
  float* ws  = (float*)d_ws;
  float* out = (float*)d_out;

  hipMemsetAsync(ws, 0, 2 * sizeof(float), stream);

  crf_forward_kernel<<<BB / 16, 32, 0, stream>>>(feats, trans, mask, ws);
  crf_gold_kernel<<<BB, 256, 0, stream>>>(feats, trans, mask, tags, ws);
  crf_finalize_kernel<<<1, 1, 0, stream>>>(ws, out);

  (void)in_sizes; (void)n_in; (void)out_size; (void)ws_size;
}